// DaRnnDecoder_76845554860771
// MI455X (gfx1250) — compile-verified
//
#include <hip/hip_runtime.h>
#include <hip/hip_bf16.h>
#include <math.h>

#define B_    256
#define TM1_  511
#define E_    128
#define D_    128
#define ATTN_ 384
#define G4D_  512

typedef __attribute__((ext_vector_type(2))) float v2f;
typedef __attribute__((ext_vector_type(8))) float v8f;

__device__ __forceinline__ float sigm_(float x) { return 1.0f / (1.0f + __expf(-x)); }

// ---------------------------------------------------------------------------
// K0 (once): Xproj[b*TM1+t, e] = sum_k X[b,t,k] * W1[e, 2D+k] + b1[e]
// M = B*TM1 = 130816 rows, N = E = 128, K = E = 128.  One wave per 16x16 tile.
// ---------------------------------------------------------------------------
__global__ __launch_bounds__(256) void k_xproj(
    const float* __restrict__ X, const float* __restrict__ W1,
    const float* __restrict__ b1, float* __restrict__ Xproj) {
  const int wave  = (blockIdx.x * blockDim.x + threadIdx.x) >> 5;
  const int lane  = threadIdx.x & 31;
  const int mtile = wave >> 3;   // 0..8175
  const int ntile = wave & 7;    // 0..7
  const int m0 = mtile << 4;
  const int n0 = ntile << 4;
  const int lr = lane & 15;
  const int hi = lane >> 4;
  const float* __restrict__ arow = X  + (size_t)(m0 + lr) * E_;
  const float* __restrict__ brow = W1 + (size_t)(n0 + lr) * ATTN_ + 2 * D_;
  v8f acc = {0.f, 0.f, 0.f, 0.f, 0.f, 0.f, 0.f, 0.f};
#pragma unroll 8
  for (int k = 0; k < E_; k += 4) {
    v2f a, b;
    // A 16x4 f32: lane L holds M=L&15; elem j -> K = k + hi*2 + j
    a.x = arow[k + hi * 2 + 0];
    a.y = arow[k + hi * 2 + 1];
    // B 4x16 f32: lane L holds N=L&15; elem j -> K = k + hi*2 + j
    b.x = brow[k + hi * 2 + 0];
    b.y = brow[k + hi * 2 + 1];
    acc = __builtin_amdgcn_wmma_f32_16x16x4_f32(false, a, false, b,
                                                (short)0, acc, false, false);
  }
  const float bias = b1[n0 + lr];
#pragma unroll
  for (int i = 0; i < 8; ++i)
    Xproj[(size_t)(m0 + i + hi * 8) * E_ + (n0 + lr)] = acc[i] + bias;
}

// ---------------------------------------------------------------------------
// K1 (per step): hcproj[b,e] = sum_{k<256} s[b,k] * W1[e,k]   (s = [h|c])
// M = 256, N = 128, K = 256.  128 tiles -> 16 blocks of 8 waves.
// ---------------------------------------------------------------------------
__global__ __launch_bounds__(256) void k_hcproj(
    const float* __restrict__ s, const float* __restrict__ W1,
    float* __restrict__ hcproj) {
  const int wave  = (blockIdx.x * blockDim.x + threadIdx.x) >> 5;
  const int lane  = threadIdx.x & 31;
  const int mtile = wave >> 3;   // 0..15
  const int ntile = wave & 7;    // 0..7
  const int m0 = mtile << 4;
  const int n0 = ntile << 4;
  const int lr = lane & 15;
  const int hi = lane >> 4;
  const float* __restrict__ arow = s  + (size_t)(m0 + lr) * (2 * D_);
  const float* __restrict__ brow = W1 + (size_t)(n0 + lr) * ATTN_;
  v8f acc = {0.f, 0.f, 0.f, 0.f, 0.f, 0.f, 0.f, 0.f};
#pragma unroll 8
  for (int k = 0; k < 2 * D_; k += 4) {
    v2f a, b;
    a.x = arow[k + hi * 2 + 0];
    a.y = arow[k + hi * 2 + 1];
    b.x = brow[k + hi * 2 + 0];
    b.y = brow[k + hi * 2 + 1];
    acc = __builtin_amdgcn_wmma_f32_16x16x4_f32(false, a, false, b,
                                                (short)0, acc, false, false);
  }
#pragma unroll
  for (int i = 0; i < 8; ++i)
    hcproj[(size_t)(m0 + i + hi * 8) * E_ + (n0 + lr)] = acc[i];
}

// ---------------------------------------------------------------------------
// K2 (per step): aw[tt*B + b] = b2 + sum_e W2[e]*tanh(Xproj[b,tt,e]+hcproj[b,e])
// One wave per (b,tt) row; float4 per lane (e = lane*4..lane*4+3) so the whole
// 512B row is one global_load_b128 per operand; shfl-xor reduction.
// ---------------------------------------------------------------------------
__global__ __launch_bounds__(256) void k_attn(
    const float* __restrict__ Xproj, const float* __restrict__ hcproj,
    const float* __restrict__ W2, const float* __restrict__ b2,
    float* __restrict__ aw) {
  const int gwave = (blockIdx.x * blockDim.x + threadIdx.x) >> 5;  // 0..130815
  const int lane  = threadIdx.x & 31;
  const int b  = gwave / TM1_;
  const int tt = gwave - b * TM1_;
  const float4 x4 = ((const float4*)(Xproj + (size_t)gwave * E_))[lane];
  const float4 h4 = ((const float4*)(hcproj + (size_t)b * E_))[lane];
  const float4 w4 = ((const float4*)W2)[lane];
  float partial = w4.x * tanhf(x4.x + h4.x)
                + w4.y * tanhf(x4.y + h4.y)
                + w4.z * tanhf(x4.z + h4.z)
                + w4.w * tanhf(x4.w + h4.w);
#pragma unroll
  for (int off = 16; off > 0; off >>= 1)
    partial += __shfl_xor(partial, off, 32);
  if (lane == 0) aw[(size_t)tt * B_ + b] = partial + b2[0];
}

// ---------------------------------------------------------------------------
// K3 (per step): softmax over the BATCH dim (faithful to reference dim=0).
// One block per tt; 256 threads = one per b.  In-place on aw.
// ---------------------------------------------------------------------------
__global__ __launch_bounds__(256) void k_softmax(float* __restrict__ aw) {
  __shared__ float red[256];
  const int tt = blockIdx.x;
  const int b  = threadIdx.x;
  const float x = aw[(size_t)tt * B_ + b];
  red[b] = x; __syncthreads();
#pragma unroll
  for (int s = 128; s > 0; s >>= 1) {
    if (b < s) red[b] = fmaxf(red[b], red[b + s]);
    __syncthreads();
  }
  const float m = red[0]; __syncthreads();
  const float e = __expf(x - m);
  red[b] = e; __syncthreads();
#pragma unroll
  for (int s = 128; s > 0; s >>= 1) {
    if (b < s) red[b] += red[b + s];
    __syncthreads();
  }
  aw[(size_t)tt * B_ + b] = e / red[0];
}

// ---------------------------------------------------------------------------
// K4 (per step): ctx[b,e] = sum_tt aw[tt,b] * X[b,tt,e];
//                ty[b] = fc_b + fc_W[0:E].ctx[b] + fc_W[E]*y_hist[b,st]
// 4 batches per 128-thread block (one wave per batch); float4 per lane so the
// whole 512B row is one global_load_b128 per tt; shfl reduction for fc dot.
// ---------------------------------------------------------------------------
__global__ __launch_bounds__(128) void k_ctx(
    const float* __restrict__ X, const float* __restrict__ aw,
    const float* __restrict__ y_hist, const float* __restrict__ fc_W,
    const float* __restrict__ fc_b, float* __restrict__ ctx,
    float* __restrict__ ty, int st) {
  const int lane = threadIdx.x & 31;
  const int sub  = threadIdx.x >> 5;
  const int b    = blockIdx.x * 4 + sub;
  const float4* __restrict__ xb =
      ((const float4*)(X + (size_t)b * TM1_ * E_)) + lane;  // row stride 32 f4
  float4 acc = {0.f, 0.f, 0.f, 0.f};
#pragma unroll 4
  for (int tt = 0; tt < TM1_; ++tt) {
    const float w = aw[(size_t)tt * B_ + b];
    const float4 x = xb[(size_t)tt * 32];
    acc.x += w * x.x; acc.y += w * x.y; acc.z += w * x.z; acc.w += w * x.w;
  }
  ((float4*)(ctx + (size_t)b * E_))[lane] = acc;
  const float4 fw = ((const float4*)fc_W)[lane];
  float partial = fw.x * acc.x + fw.y * acc.y + fw.z * acc.z + fw.w * acc.w;
#pragma unroll
  for (int off = 16; off > 0; off >>= 1)
    partial += __shfl_xor(partial, off, 32);
  if (lane == 0)
    ty[b] = partial + fc_W[E_] * y_hist[(size_t)b * TM1_ + st] + fc_b[0];
}

// ---------------------------------------------------------------------------
// K5 (per step): gates[b,n] = ty[b]*W_ih[n] + b_ih[n] + b_hh[n]
//                           + sum_k h[b,k]*W_hh[n,k]
// M = 256, N = 512, K = 128.  512 tiles -> 64 blocks of 8 waves.
// ---------------------------------------------------------------------------
__global__ __launch_bounds__(256) void k_gates(
    const float* __restrict__ s, const float* __restrict__ W_hh,
    const float* __restrict__ W_ih, const float* __restrict__ b_ih,
    const float* __restrict__ b_hh, const float* __restrict__ ty,
    float* __restrict__ gates) {
  const int wave  = (blockIdx.x * blockDim.x + threadIdx.x) >> 5;
  const int lane  = threadIdx.x & 31;
  const int mtile = wave >> 5;   // 0..15
  const int ntile = wave & 31;   // 0..31
  const int m0 = mtile << 4;
  const int n0 = ntile << 4;
  const int lr = lane & 15;
  const int hi = lane >> 4;
  const float* __restrict__ arow = s    + (size_t)(m0 + lr) * (2 * D_); // h part
  const float* __restrict__ brow = W_hh + (size_t)(n0 + lr) * D_;
  v8f acc = {0.f, 0.f, 0.f, 0.f, 0.f, 0.f, 0.f, 0.f};
#pragma unroll 8
  for (int k = 0; k < D_; k += 4) {
    v2f a, b;
    a.x = arow[k + hi * 2 + 0];
    a.y = arow[k + hi * 2 + 1];
    b.x = brow[k + hi * 2 + 0];
    b.y = brow[k + hi * 2 + 1];
    acc = __builtin_amdgcn_wmma_f32_16x16x4_f32(false, a, false, b,
                                                (short)0, acc, false, false);
  }
  const int n = n0 + lr;
  const float wih  = W_ih[n];
  const float bias = b_ih[n] + b_hh[n];
#pragma unroll
  for (int i = 0; i < 8; ++i) {
    const int bb = m0 + i + hi * 8;
    gates[(size_t)bb * G4D_ + n] = acc[i] + ty[bb] * wih + bias;
  }
}

// ---------------------------------------------------------------------------
// K6 (per step): LSTM pointwise update on s = [h|c].
// ---------------------------------------------------------------------------
__global__ __launch_bounds__(256) void k_update(
    const float* __restrict__ gates, float* __restrict__ s) {
  const int idx = blockIdx.x * blockDim.x + threadIdx.x;  // 0..32767
  const int b = idx >> 7;
  const int d = idx & 127;
  const float* __restrict__ g = gates + (size_t)b * G4D_;
  const float ig = g[d], fg = g[D_ + d], gg = g[2 * D_ + d], og = g[3 * D_ + d];
  const float c  = s[(size_t)b * (2 * D_) + D_ + d];
  const float c2 = sigm_(fg) * c + sigm_(ig) * tanhf(gg);
  const float h2 = sigm_(og) * tanhf(c2);
  s[(size_t)b * (2 * D_) + d]       = h2;
  s[(size_t)b * (2 * D_) + D_ + d]  = c2;
}

// ---------------------------------------------------------------------------
// K7 (once): out[b] = fcf_b + fcf_W[0:D].h[b] + fcf_W[D:D+E].ctx_last[b]
// ---------------------------------------------------------------------------
__global__ __launch_bounds__(256) void k_final(
    const float* __restrict__ s, const float* __restrict__ ctx,
    const float* __restrict__ fcf_W, const float* __restrict__ fcf_b,
    float* __restrict__ out) {
  const int b = blockIdx.x;
  const int tid = threadIdx.x;  // 0..255
  float v;
  if (tid < D_) v = fcf_W[tid] * s[(size_t)b * (2 * D_) + tid];
  else          v = fcf_W[tid] * ctx[(size_t)b * E_ + (tid - D_)];
  __shared__ float red[256];
  red[tid] = v; __syncthreads();
#pragma unroll
  for (int ss = 128; ss > 0; ss >>= 1) {
    if (tid < ss) red[tid] += red[tid + ss];
    __syncthreads();
  }
  if (tid == 0) out[b] = red[0] + fcf_b[0];
}

// ---------------------------------------------------------------------------
extern "C" void kernel_launch(void* const* d_in, const int* in_sizes, int n_in,
                              void* d_out, int out_size, void* d_ws, size_t ws_size,
                              hipStream_t stream) {
  const float* X      = (const float*)d_in[0];   // (B, TM1, E)
  const float* y_hist = (const float*)d_in[1];   // (B, TM1)
  const float* W1     = (const float*)d_in[2];   // (E, ATTN)
  const float* b1     = (const float*)d_in[3];   // (E,)
  const float* W2     = (const float*)d_in[4];   // (1, E)
  const float* b2     = (const float*)d_in[5];   // (1,)
  const float* W_ih   = (const float*)d_in[6];   // (4D, 1)
  const float* W_hh   = (const float*)d_in[7];   // (4D, D)
  const float* b_ih   = (const float*)d_in[8];   // (4D,)
  const float* b_hh   = (const float*)d_in[9];   // (4D,)
  const float* fc_W   = (const float*)d_in[10];  // (1, E+1)
  const float* fc_b   = (const float*)d_in[11];  // (1,)
  const float* fcf_W  = (const float*)d_in[12];  // (1, D+E)
  const float* fcf_b  = (const float*)d_in[13];  // (1,)

  float* ws = (float*)d_ws;
  size_t off = 0;
  float* Xproj  = ws + off; off += (size_t)B_ * TM1_ * E_;  // 16,744,448
  float* s      = ws + off; off += (size_t)B_ * 2 * D_;     // 65,536
  float* hcproj = ws + off; off += (size_t)B_ * E_;         // 32,768
  float* aw     = ws + off; off += (size_t)TM1_ * B_;       // 130,816
  float* ctx    = ws + off; off += (size_t)B_ * E_;         // 32,768
  float* ty     = ws + off; off += (size_t)B_;              // 256
  float* gates  = ws + off; off += (size_t)B_ * G4D_;       // 131,072

  // zero-init [h|c] state (stream memset is graph-capture safe)
  hipMemsetAsync(s, 0, (size_t)B_ * 2 * D_ * sizeof(float), stream);

  // One-time: Xproj = X @ W1_x.T + b1   (8176 m-tiles x 8 n-tiles, 8 waves/blk)
  k_xproj<<<8176, 256, 0, stream>>>(X, W1, b1, Xproj);

  for (int st = 0; st < TM1_; ++st) {
    k_hcproj<<<16, 256, 0, stream>>>(s, W1, hcproj);
    k_attn<<<16352, 256, 0, stream>>>(Xproj, hcproj, W2, b2, aw);
    k_softmax<<<TM1_, 256, 0, stream>>>(aw);
    k_ctx<<<64, 128, 0, stream>>>(X, aw, y_hist, fc_W, fc_b, ctx, ty, st);
    k_gates<<<64, 256, 0, stream>>>(s, W_hh, W_ih, b_ih, b_hh, ty, gates);
    k_update<<<128, 256, 0, stream>>>(gates, s);
  }

  k_final<<<B_, 256, 0, stream>>>(s, ctx, fcf_W, fcf_b, (float*)d_out);
}